// VoxelFeatureExtractor_13134009991800
// MI455X (gfx1250) — compile-verified
//
#include <hip/hip_runtime.h>
#include <math.h>

#define KVOX 30000
#define TPTS 48
#define WPB  8
#define NTHREADS (WPB * 32)
#define EPSB 1e-5f

typedef float v8f __attribute__((ext_vector_type(8)));
typedef float v2f __attribute__((ext_vector_type(2)));

// ---- dynamic LDS layout (float offsets) ----
// Weights stored pair-interleaved for single-ds_load_b64 B fragments:
//   wXf[p*STRIDE + 2*n + {0,1}] = W[2p][n], W[2p+1][n]   (p = k/2)
// STRIDE chosen == 32 mod 64 floats so lane halves (p vs p+1) hit disjoint bank halves.
#define W1F_STRIDE 32                      // 2*16, already 32 mod 64
#define W2F_STRIDE 160                     // 2*64 + 32
#define W3F_STRIDE 288                     // 2*128 + 32
#define OFF_W1F  0
#define OFF_W2F  (OFF_W1F + 4 * W1F_STRIDE)    // 128
#define OFF_W3F  (OFF_W2F + 16 * W2F_STRIDE)   // 2688
#define OFF_SC1  (OFF_W3F + 64 * W3F_STRIDE)   // 21120
#define OFF_SH1  (OFF_SC1 + 16)
#define OFF_SC2  (OFF_SH1 + 16)
#define OFF_SH2  (OFF_SC2 + 64)
#define OFF_SC3  (OFF_SH2 + 64)
#define OFF_SH3  (OFF_SC3 + 128)
#define OFF_WAVE (OFF_SH3 + 128)               // 21536

// per-wave activation scratch: even strides, gcd(stride,64) small -> conflict-free A loads
#define X0_STRIDE 12
#define H1_STRIDE 20
#define H2_STRIDE 68
#define WV_X0 0
#define WV_H1 (WV_X0 + TPTS * X0_STRIDE)  // 576
#define WV_H2 (WV_H1 + TPTS * H1_STRIDE)  // 1536
#define WV_A1 (WV_H2 + TPTS * H2_STRIDE)  // 4800
#define WV_A2 (WV_A1 + 16)                // 4816
#define WV_SIZE 4896
#define SMEM_FLOATS (OFF_WAVE + WPB * WV_SIZE)   // 60704 floats = 242816 B (< 320KB WGP LDS)

static __device__ __forceinline__ v8f wmma4(v2f a, v2f b, v8f c) {
  // V_WMMA_F32_16X16X4_F32: D = A(16x4) x B(4x16) + C(16x16), fp32 matrix pipe
  return __builtin_amdgcn_wmma_f32_16x16x4_f32(false, a, false, b, (short)0, c, false, false);
}

__global__ void __launch_bounds__(NTHREADS, 1)
vfe_wmma_kernel(const float* __restrict__ feat, const int* __restrict__ numv,
                const float* __restrict__ w1, const float* __restrict__ b1,
                const float* __restrict__ g1, const float* __restrict__ be1,
                const float* __restrict__ m1, const float* __restrict__ var1,
                const float* __restrict__ w2, const float* __restrict__ b2,
                const float* __restrict__ g2, const float* __restrict__ be2,
                const float* __restrict__ m2, const float* __restrict__ var2,
                const float* __restrict__ w3, const float* __restrict__ b3,
                const float* __restrict__ g3, const float* __restrict__ be3,
                const float* __restrict__ m3, const float* __restrict__ var3,
                float* __restrict__ out)
{
  extern __shared__ float smem[];
  const int tid = threadIdx.x;

  // ---- cooperative weight staging: pair-interleaved B-fragment layout ----
  for (int i = tid; i < 4 * 16; i += NTHREADS) {          // w1: 4 pairs x 16 cols
    int p = i >> 4, n = i & 15;
    smem[OFF_W1F + p * W1F_STRIDE + 2 * n + 0] = w1[(2 * p + 0) * 16 + n];
    smem[OFF_W1F + p * W1F_STRIDE + 2 * n + 1] = w1[(2 * p + 1) * 16 + n];
  }
  for (int i = tid; i < 16 * 64; i += NTHREADS) {         // w2: 16 pairs x 64 cols
    int p = i >> 6, n = i & 63;
    smem[OFF_W2F + p * W2F_STRIDE + 2 * n + 0] = w2[(2 * p + 0) * 64 + n];
    smem[OFF_W2F + p * W2F_STRIDE + 2 * n + 1] = w2[(2 * p + 1) * 64 + n];
  }
  for (int i = tid; i < 64 * 128; i += NTHREADS) {        // w3: 64 pairs x 128 cols
    int p = i >> 7, n = i & 127;
    smem[OFF_W3F + p * W3F_STRIDE + 2 * n + 0] = w3[(2 * p + 0) * 128 + n];
    smem[OFF_W3F + p * W3F_STRIDE + 2 * n + 1] = w3[(2 * p + 1) * 128 + n];
  }
  for (int i = tid; i < 16; i += NTHREADS) {
    float sc = g1[i] / sqrtf(var1[i] + EPSB);
    smem[OFF_SC1 + i] = sc;
    smem[OFF_SH1 + i] = (b1[i] - m1[i]) * sc + be1[i];
  }
  for (int i = tid; i < 64; i += NTHREADS) {
    float sc = g2[i] / sqrtf(var2[i] + EPSB);
    smem[OFF_SC2 + i] = sc;
    smem[OFF_SH2 + i] = (b2[i] - m2[i]) * sc + be2[i];
  }
  for (int i = tid; i < 128; i += NTHREADS) {
    float sc = g3[i] / sqrtf(var3[i] + EPSB);
    smem[OFF_SC3 + i] = sc;
    smem[OFF_SH3 + i] = (b3[i] - m3[i]) * sc + be3[i];
  }
  __syncthreads();

  const int lane = tid & 31;
  const int wave = tid >> 5;
  const int v = blockIdx.x * WPB + wave;     // wave-uniform
  if (v >= KVOX) return;

  float* xw  = smem + OFF_WAVE + wave * WV_SIZE;
  float* x0  = xw + WV_X0;
  float* h1  = xw + WV_H1;
  float* h2  = xw + WV_H2;
  float* a1s = xw + WV_A1;
  float* a2s = xw + WV_A2;

  const int nv = numv[v];
  const float* fv = feat + (size_t)v * (TPTS * 4);

  // ---- stage 0: mean / rel / dist / mask -> x0 (48 x 8) ----
  float4 fA = ((const float4*)fv)[lane];                       // row t = lane
  float4 fB = make_float4(0.f, 0.f, 0.f, 0.f);
  if (lane < 16) fB = ((const float4*)fv)[lane + 32];          // row t = lane + 32

  float sx = fA.x + fB.x, sy = fA.y + fB.y, sz = fA.z + fB.z;  // mean over ALL 48 rows
  #pragma unroll
  for (int off = 16; off >= 1; off >>= 1) {
    sx += __shfl_xor(sx, off, 32);
    sy += __shfl_xor(sy, off, 32);
    sz += __shfl_xor(sz, off, 32);
  }
  const float inv = 1.0f / (float)nv;
  const float mxm = sx * inv, mym = sy * inv, mzm = sz * inv;

  {
    int t = lane;
    float msk = (t < nv) ? 1.f : 0.f;
    float dist = sqrtf(fA.x * fA.x + fA.y * fA.y + fA.z * fA.z);
    float* r = x0 + t * X0_STRIDE;
    ((float4*)r)[0] = make_float4(fA.x * msk, fA.y * msk, fA.z * msk, fA.w * msk);
    ((float4*)r)[1] = make_float4((fA.x - mxm) * msk, (fA.y - mym) * msk,
                                  (fA.z - mzm) * msk, dist * msk);
  }
  if (lane < 16) {
    int t = lane + 32;
    float msk = (t < nv) ? 1.f : 0.f;
    float dist = sqrtf(fB.x * fB.x + fB.y * fB.y + fB.z * fB.z);
    float* r = x0 + t * X0_STRIDE;
    ((float4*)r)[0] = make_float4(fB.x * msk, fB.y * msk, fB.z * msk, fB.w * msk);
    ((float4*)r)[1] = make_float4((fB.x - mxm) * msk, (fB.y - mym) * msk,
                                  (fB.z - mzm) * msk, dist * msk);
  }

  const int lm   = lane & 15;
  const int kbh  = (lane < 16) ? 0 : 1;   // pair index offset: p = 2*ks + kbh
  const int kb   = kbh * 2;               // K offset within fragment for A loads
  const int rsel = (lane < 16) ? 0 : 8;   // D rows: lanes 0-15 -> M=r, 16-31 -> M=8+r
  const v8f vzero = {0.f, 0.f, 0.f, 0.f, 0.f, 0.f, 0.f, 0.f};

  // per-lane fragment base pointers into the pair-interleaved weights
  const float* bw1 = smem + OFF_W1F + kbh * W1F_STRIDE + 2 * lm;
  const float* bw2 = smem + OFF_W2F + kbh * W2F_STRIDE + 2 * lm;
  const float* bw3 = smem + OFF_W3F + kbh * W3F_STRIDE + 2 * lm;

  // ---- stage 1: 48x8 @ 8x16 -> BN -> ReLU -> h1 (masked), agg1 ----
  float rmax1 = 0.f;                       // relu >= 0 so 0 is a safe identity
  const float sc1v = smem[OFF_SC1 + lm];
  const float sh1v = smem[OFF_SH1 + lm];
  #pragma unroll
  for (int mt = 0; mt < 3; ++mt) {
    const int m = lm + 16 * mt;
    v2f a0 = *(const v2f*)(x0 + m * X0_STRIDE + 0 + kb);
    v2f a1 = *(const v2f*)(x0 + m * X0_STRIDE + 4 + kb);
    v2f b0 = *(const v2f*)(bw1 + 0 * W1F_STRIDE);   // p = 2*0 + kbh
    v2f b1v = *(const v2f*)(bw1 + 2 * W1F_STRIDE);  // p = 2*1 + kbh
    v8f acc = vzero;
    acc = wmma4(a0, b0, acc);
    acc = wmma4(a1, b1v, acc);
    #pragma unroll
    for (int r = 0; r < 8; ++r) {
      float h = fmaxf(fmaf(acc[r], sc1v, sh1v), 0.f); // rows t>=nv give relu(shift) = ref behavior
      int t = 16 * mt + rsel + r;
      h1[t * H1_STRIDE + lm] = (t < nv) ? h : 0.f;
      rmax1 = fmaxf(rmax1, h);
    }
  }
  rmax1 = fmaxf(rmax1, __shfl_xor(rmax1, 16, 32));
  if (lane < 16) a1s[lane] = rmax1;

  // ---- stage 2: 48x32 @ 32x64 -> BN -> ReLU -> h2 (masked), agg2 ----
  float rmax2[4] = {0.f, 0.f, 0.f, 0.f};
  #pragma unroll 1
  for (int mt = 0; mt < 3; ++mt) {
    const int m = lm + 16 * mt;
    const float mskA = (m < nv) ? 1.f : 0.f;
    v8f acc[4];
    #pragma unroll
    for (int nt = 0; nt < 4; ++nt) acc[nt] = vzero;
    #pragma unroll
    for (int ks = 0; ks < 8; ++ks) {
      v2f a;
      if (ks < 4) {
        a = *(const v2f*)(h1 + m * H1_STRIDE + 4 * ks + kb);
      } else {                                     // broadcast-agg half of concat, masked
        v2f g = *(const v2f*)(a1s + 4 * ks + kb - 16);
        a.x = g.x * mskA; a.y = g.y * mskA;
      }
      const float* bp = bw2 + (2 * ks) * W2F_STRIDE;
      #pragma unroll
      for (int nt = 0; nt < 4; ++nt) {
        v2f b = *(const v2f*)(bp + 32 * nt);       // one ds_load_b64 per B fragment
        acc[nt] = wmma4(a, b, acc[nt]);
      }
    }
    #pragma unroll
    for (int nt = 0; nt < 4; ++nt) {
      const int n = 16 * nt + lm;
      const float sc = smem[OFF_SC2 + n], sh = smem[OFF_SH2 + n];
      #pragma unroll
      for (int r = 0; r < 8; ++r) {
        float h = fmaxf(fmaf(acc[nt][r], sc, sh), 0.f);
        int t = 16 * mt + rsel + r;
        h2[t * H2_STRIDE + n] = (t < nv) ? h : 0.f;
        rmax2[nt] = fmaxf(rmax2[nt], h);
      }
    }
  }
  #pragma unroll
  for (int nt = 0; nt < 4; ++nt) {
    float mx = fmaxf(rmax2[nt], __shfl_xor(rmax2[nt], 16, 32));
    if (lane < 16) a2s[16 * nt + lane] = mx;
  }

  // ---- stage 3: 48x128 @ 128x128 -> BN -> ReLU -> mask -> max over T ----
  float rmax3[8] = {0.f, 0.f, 0.f, 0.f, 0.f, 0.f, 0.f, 0.f};
  #pragma unroll 1
  for (int mt = 0; mt < 3; ++mt) {
    const int m = lm + 16 * mt;
    const float mskA = (m < nv) ? 1.f : 0.f;
    v8f acc[8];
    #pragma unroll
    for (int nt = 0; nt < 8; ++nt) acc[nt] = vzero;
    #pragma unroll 4
    for (int ks = 0; ks < 32; ++ks) {
      v2f a;
      if (ks < 16) {
        a = *(const v2f*)(h2 + m * H2_STRIDE + 4 * ks + kb);
      } else {
        v2f g = *(const v2f*)(a2s + 4 * ks + kb - 64);
        a.x = g.x * mskA; a.y = g.y * mskA;
      }
      const float* bp = bw3 + (2 * ks) * W3F_STRIDE;
      #pragma unroll
      for (int nt = 0; nt < 8; ++nt) {
        v2f b = *(const v2f*)(bp + 32 * nt);       // one ds_load_b64 per B fragment
        acc[nt] = wmma4(a, b, acc[nt]);
      }
    }
    #pragma unroll
    for (int nt = 0; nt < 8; ++nt) {
      const int n = 16 * nt + lm;
      const float sc = smem[OFF_SC3 + n], sh = smem[OFF_SH3 + n];
      #pragma unroll
      for (int r = 0; r < 8; ++r) {
        float h = fmaxf(fmaf(acc[nt][r], sc, sh), 0.f);
        int t = 16 * mt + rsel + r;
        rmax3[nt] = fmaxf(rmax3[nt], (t < nv) ? h : 0.f);   // final mask before max
      }
    }
  }
  #pragma unroll
  for (int nt = 0; nt < 8; ++nt) {
    float mx = fmaxf(rmax3[nt], __shfl_xor(rmax3[nt], 16, 32));
    if (lane < 16) out[(size_t)v * 128 + 16 * nt + lane] = mx;
  }
}

extern "C" void kernel_launch(void* const* d_in, const int* in_sizes, int n_in,
                              void* d_out, int out_size, void* d_ws, size_t ws_size,
                              hipStream_t stream) {
  (void)in_sizes; (void)n_in; (void)out_size; (void)d_ws; (void)ws_size;
  const float* feat = (const float*)d_in[0];
  const int*   numv = (const int*)d_in[1];
  // d_in[2] = coors (unused by reference output)
  const float* w1  = (const float*)d_in[3];
  const float* b1  = (const float*)d_in[4];
  const float* g1  = (const float*)d_in[5];
  const float* be1 = (const float*)d_in[6];
  const float* m1  = (const float*)d_in[7];
  const float* vv1 = (const float*)d_in[8];
  const float* w2  = (const float*)d_in[9];
  const float* b2  = (const float*)d_in[10];
  const float* g2  = (const float*)d_in[11];
  const float* be2 = (const float*)d_in[12];
  const float* m2  = (const float*)d_in[13];
  const float* vv2 = (const float*)d_in[14];
  const float* w3  = (const float*)d_in[15];
  const float* b3  = (const float*)d_in[16];
  const float* g3  = (const float*)d_in[17];
  const float* be3 = (const float*)d_in[18];
  const float* m3  = (const float*)d_in[19];
  const float* vv3 = (const float*)d_in[20];
  float* out = (float*)d_out;

  dim3 grid((KVOX + WPB - 1) / WPB);   // 3750 blocks, one voxel per wave, 8 waves fill a WGP
  dim3 block(NTHREADS);
  size_t shmem = (size_t)SMEM_FLOATS * sizeof(float);   // ~237 KB (CDNA5 WGP: 320 KB LDS)
  vfe_wmma_kernel<<<grid, block, shmem, stream>>>(
      feat, numv, w1, b1, g1, be1, m1, vv1,
      w2, b2, g2, be2, m2, vv2, w3, b3, g3, be3, m3, vv3, out);
}